// LanguageModel_38551626449183
// MI455X (gfx1250) — compile-verified
//
#include <hip/hip_runtime.h>

// GRU language model for MI455X (gfx1250), wave32 + WMMA f32 16x16x4.
//
// Persistent-block design: each of 64 blocks owns 16 batch rows for the
// entire 512-step scan. Weights live in LDS pair-interleaved ([K/2][NP][2])
// so every WMMA B operand is one aligned ds_load_b64; A pitches are even so
// A operands are aligned ds_load_b64 too (no operand-shuffle movs).
// Per step:
//   xp = emb @ W_ih^T + b_ih   (K=50 padded to 52 -> 13 WMMA k-steps)
//   hp = h   @ W_hh^T + b_hh   (K=100            -> 25 WMMA k-steps)
// via V_WMMA_F32_16X16X4_F32, then a VALU gate-fusion phase updates h.
// Embedding gathers for step t+1 are issued before step t's WMMA phase and
// committed to LDS at the top of the next step (latency hidden by compute).

typedef __attribute__((ext_vector_type(2))) float v2f;
typedef __attribute__((ext_vector_type(8))) float v8f;

#define EMB     50
#define HID     100
#define GATES   300            // 3*HID
#define NP      304            // padded gate dim: 19 tiles * 16
#define NTILES  19
#define KE      52             // padded emb K: 13 tiles * 4
#define EPITCH  58             // even (b64-aligned A pairs), distinct bank rows
#define HPITCH  102            // even (b64-aligned A pairs), distinct bank rows
#define BATCH   1024
#define SEQ     512
#define NTHREADS 256           // 8 waves of 32
#define GELEMS  (16 * EMB)     // 800 gathered embedding elements per step
#define GPT     4              // ceil(800 / 256) gather values per thread

// LDS plan (dwords)
#define OFF_WIH   0                              // (KE/2)*NP*2 = 15808
#define OFF_WHH   (OFF_WIH + KE * NP)            // (HID/2)*NP*2 = 30400
#define OFF_XP    (OFF_WHH + HID * NP)           // 16*NP = 4864
#define OFF_HP    (OFF_XP + 16 * NP)             // 16*NP = 4864
#define OFF_H     (OFF_HP + 16 * NP)             // 16*HPITCH = 1632
#define OFF_EMB   (OFF_H + 16 * HPITCH)          // 16*EPITCH = 928
#define OFF_BIH   (OFF_EMB + 16 * EPITCH)        // NP
#define OFF_BHH   (OFF_BIH + NP)                 // NP
#define OFF_TOK   (OFF_BHH + NP)                 // 16*SEQ ints = 8192
#define LDS_DWORDS (OFF_TOK + 16 * SEQ)          // 67296 dw = 269,184 B (<320 KB)

__device__ __forceinline__ float sigmoidf_(float x) {
    return 1.0f / (1.0f + __expf(-x));
}

#if __has_builtin(__builtin_amdgcn_tanhf)
#define FAST_TANH(x) __builtin_amdgcn_tanhf(x)
#else
#define FAST_TANH(x) tanhf(x)
#endif

extern "C" __global__ __launch_bounds__(NTHREADS)
void gru_lm_kernel(const int*   __restrict__ x,
                   const float* __restrict__ emb_table,
                   const float* __restrict__ W_ih,
                   const float* __restrict__ W_hh,
                   const float* __restrict__ b_ih,
                   const float* __restrict__ b_hh,
                   const float* __restrict__ W_fc,
                   const float* __restrict__ b_fc,
                   float*       __restrict__ out)
{
    extern __shared__ float smem[];
    float* WihP  = smem + OFF_WIH;   // [KE/2][NP][2]  pair-interleaved W_ih^T
    float* WhhP  = smem + OFF_WHH;   // [HID/2][NP][2] pair-interleaved W_hh^T
    float* xp_s  = smem + OFF_XP;    // [16][NP]
    float* hp_s  = smem + OFF_HP;    // [16][NP]
    float* h_s   = smem + OFF_H;     // [16][HPITCH]
    float* emb_s = smem + OFF_EMB;   // [16][EPITCH]
    float* bih_s = smem + OFF_BIH;   // [NP]
    float* bhh_s = smem + OFF_BHH;   // [NP]
    int*   tok_s = (int*)(smem + OFF_TOK);  // [16][SEQ]

    const int tid   = threadIdx.x;
    const int wave  = tid >> 5;
    const int lane  = tid & 31;
    const int lrow  = lane & 15;   // row (A) / column (B,C,D) within 16x16 tile
    const int lhi   = lane >> 4;   // half-wave select (K pairs / row halves)
    const int brow0 = blockIdx.x * 16;

    // Zero all LDS (establishes pad regions and h0 = 0).
    for (int i = tid; i < LDS_DWORDS; i += NTHREADS) smem[i] = 0.0f;
    __syncthreads();

    // Stage pair-interleaved transposed weights, biases, and all tokens.
    for (int i = tid; i < GATES * EMB; i += NTHREADS) {
        int g = i / EMB, e = i % EMB;
        WihP[((e >> 1) * NP + g) * 2 + (e & 1)] = W_ih[i];
    }
    for (int i = tid; i < GATES * HID; i += NTHREADS) {
        int g = i / HID, k = i % HID;
        WhhP[((k >> 1) * NP + g) * 2 + (k & 1)] = W_hh[i];
    }
    for (int i = tid; i < GATES; i += NTHREADS) {
        bih_s[i] = b_ih[i];
        bhh_s[i] = b_hh[i];
    }
    for (int i = tid; i < 16 * SEQ; i += NTHREADS) {
        tok_s[i] = x[brow0 * SEQ + i];     // rows are contiguous: coalesced
    }
    __syncthreads();

    // Software-pipelined embedding gather: issue loads for step t while
    // step t-1 computes; commit to LDS at the top of step t.
    float gv[GPT];
#define GATHER_ISSUE(T)                                                     \
    {                                                                       \
        _Pragma("unroll")                                                   \
        for (int u = 0; u < GPT; ++u) {                                     \
            int e = tid + u * NTHREADS;                                     \
            if (e < GELEMS) {                                               \
                int row = e / EMB, c = e % EMB;                             \
                gv[u] = emb_table[tok_s[row * SEQ + (T)] * EMB + c];        \
            }                                                               \
        }                                                                   \
    }
#define GATHER_COMMIT()                                                     \
    {                                                                       \
        _Pragma("unroll")                                                   \
        for (int u = 0; u < GPT; ++u) {                                     \
            int e = tid + u * NTHREADS;                                     \
            if (e < GELEMS) {                                               \
                int row = e / EMB, c = e % EMB;                             \
                emb_s[row * EPITCH + c] = gv[u];                            \
            }                                                               \
        }                                                                   \
    }

    GATHER_ISSUE(0);

    for (int t = 0; t < SEQ; ++t) {
        GATHER_COMMIT();                 // waits only here on the gather
        __syncthreads();
        if (t + 1 < SEQ) GATHER_ISSUE(t + 1);   // overlap with WMMA below

        // ---- WMMA projections, 19 N-tiles over 8 waves ----
        for (int nt = wave; nt < NTILES; nt += NTHREADS / 32) {
            const int n0  = nt * 16;
            const int col = n0 + lrow;
            const float bi = bih_s[col];
            const float bh = bhh_s[col];
            v8f xacc, hacc;
#pragma unroll
            for (int r = 0; r < 8; ++r) { xacc[r] = bi; hacc[r] = bh; }

            // xp += emb[16xK] * Wih^T[Kx16]
#pragma unroll
            for (int k0 = 0; k0 < KE; k0 += 4) {
                const int kb = k0 + 2 * lhi;           // even
                const int kp = (k0 >> 1) + lhi;
                v2f a = *reinterpret_cast<const v2f*>(&emb_s[lrow * EPITCH + kb]);
                v2f b = *reinterpret_cast<const v2f*>(&WihP[(kp * NP + col) * 2]);
                xacc = __builtin_amdgcn_wmma_f32_16x16x4_f32(
                    false, a, false, b, (short)0, xacc, false, false);
            }
            // hp += h[16xK] * Whh^T[Kx16]   (K = 100 = 25 exact k-steps)
#pragma unroll
            for (int k0 = 0; k0 < HID; k0 += 4) {
                const int kb = k0 + 2 * lhi;           // even
                const int kp = (k0 >> 1) + lhi;
                v2f a = *reinterpret_cast<const v2f*>(&h_s[lrow * HPITCH + kb]);
                v2f b = *reinterpret_cast<const v2f*>(&WhhP[(kp * NP + col) * 2]);
                hacc = __builtin_amdgcn_wmma_f32_16x16x4_f32(
                    false, a, false, b, (short)0, hacc, false, false);
            }
            // D layout: lanes 0-15 hold rows 0-7, lanes 16-31 rows 8-15.
            const int dr0 = lhi * 8;
#pragma unroll
            for (int r = 0; r < 8; ++r) {
                xp_s[(dr0 + r) * NP + col] = xacc[r];
                hp_s[(dr0 + r) * NP + col] = hacc[r];
            }
        }
        __syncthreads();

        // ---- Gate fusion + hidden-state update ----
        for (int e = tid; e < 16 * HID; e += NTHREADS) {
            int row = e / HID, j = e % HID;
            float rg = sigmoidf_(xp_s[row * NP + j]           + hp_s[row * NP + j]);
            float zg = sigmoidf_(xp_s[row * NP + j + HID]     + hp_s[row * NP + j + HID]);
            float ng = FAST_TANH(xp_s[row * NP + j + 2 * HID] + rg * hp_s[row * NP + j + 2 * HID]);
            float h  = h_s[row * HPITCH + j];
            h_s[row * HPITCH + j] = (1.0f - zg) * ng + zg * h;
        }
        __syncthreads();
    }

    // ---- Final FC: out[16][2] = h @ W_fc^T + b_fc ----
    if (tid < 32) {
        int row = tid >> 1, o = tid & 1;
        float acc = b_fc[o];
#pragma unroll 4
        for (int j = 0; j < HID; ++j)
            acc += h_s[row * HPITCH + j] * W_fc[o * HID + j];
        out[(brow0 + row) * 2 + o] = acc;
    }
}

extern "C" void kernel_launch(void* const* d_in, const int* in_sizes, int n_in,
                              void* d_out, int out_size, void* d_ws, size_t ws_size,
                              hipStream_t stream) {
    const int*   x         = (const int*)  d_in[0];
    const float* emb_table = (const float*)d_in[1];
    const float* W_ih      = (const float*)d_in[2];
    const float* W_hh      = (const float*)d_in[3];
    const float* b_ih      = (const float*)d_in[4];
    const float* b_hh      = (const float*)d_in[5];
    const float* W_fc      = (const float*)d_in[6];
    const float* b_fc      = (const float*)d_in[7];
    float*       out       = (float*)d_out;

    const size_t shbytes = (size_t)LDS_DWORDS * sizeof(float);  // ~269 KB
    gru_lm_kernel<<<dim3(BATCH / 16), dim3(NTHREADS), shbytes, stream>>>(
        x, emb_table, W_ih, W_hh, b_ih, b_hh, W_fc, b_fc, out);
}